// AttentiveMLP1_6236292513983
// MI455X (gfx1250) — compile-verified
//
#include <hip/hip_runtime.h>
#include <hip/hip_bf16.h>
#include <math.h>

#define N_NODES   100000
#define N_EDGES   1600000
#define EDGE_FEAT 16
#define EDGE_HID  32
#define NODE_FEAT 128
#define H_IN_DIM  (EDGE_HID + NODE_FEAT)   // 160
#define M_TILE    32                       // nodes per block (2 WMMA M-tiles)

typedef __attribute__((ext_vector_type(2))) float v2f;
typedef __attribute__((ext_vector_type(8))) float v8f;

// ---- order-preserving float <-> u32 encoding for atomic max ----
__device__ __forceinline__ unsigned fenc(float f) {
    unsigned u = __float_as_uint(f);
    return (u >> 31) ? ~u : (u | 0x80000000u);
}
__device__ __forceinline__ float fdec(unsigned e) {
    return __uint_as_float((e >> 31) ? (e ^ 0x80000000u) : ~e);
}
#define ENC_NEG_INF 0x007FFFFFu   // fenc(-inf)

// ---------------------------------------------------------------
// Kernel 1: init scratch (denom=0, c=0, seg_max=enc(-inf))
// ---------------------------------------------------------------
__global__ __launch_bounds__(256) void k_init(unsigned* __restrict__ seg_max,
                                              float* __restrict__ denom,
                                              float* __restrict__ c) {
    int i = blockIdx.x * 256 + threadIdx.x;
    if (i < N_NODES * EDGE_HID) c[i] = 0.0f;
    if (i < N_NODES) { seg_max[i] = ENC_NEG_INF; denom[i] = 0.0f; }
}

// ---------------------------------------------------------------
// Kernel 2: segment max of logits over dst
// ---------------------------------------------------------------
__global__ __launch_bounds__(256) void k_segmax(const float* __restrict__ logits,
                                                const int* __restrict__ dst,
                                                unsigned* __restrict__ seg_max,
                                                int n_edges) {
    int e = blockIdx.x * 256 + threadIdx.x;
    if (e >= n_edges) return;
    atomicMax(&seg_max[dst[e]], fenc(logits[e]));
}

// ---------------------------------------------------------------
// Kernel 3: ex = exp(logit - max), denom = segment_sum(ex)
// ---------------------------------------------------------------
__global__ __launch_bounds__(256) void k_expsum(const float* __restrict__ logits,
                                                const int* __restrict__ dst,
                                                const unsigned* __restrict__ seg_max,
                                                float* __restrict__ exbuf,
                                                float* __restrict__ denom,
                                                int n_edges) {
    int e = blockIdx.x * 256 + threadIdx.x;
    if (e >= n_edges) return;
    int d = dst[e];
    float m = fdec(seg_max[d]);
    if (!isfinite(m)) m = 0.0f;
    float x = expf(logits[e] - m);
    exbuf[e] = x;
    atomicAdd(&denom[d], x);
}

// ---------------------------------------------------------------
// Kernel 4: e = alpha * (edge_feats @ W_et + b_et); c = segsum(e)
// W_et (16x32, 2KB) staged in LDS; one thread per edge.
// ---------------------------------------------------------------
__global__ __launch_bounds__(256) void k_edge_xform(const float* __restrict__ edge_feats,
                                                    const float* __restrict__ exbuf,
                                                    const float* __restrict__ denom,
                                                    const int* __restrict__ dst,
                                                    const float* __restrict__ W_et,
                                                    const float* __restrict__ b_et,
                                                    float* __restrict__ c,
                                                    int n_edges) {
    __shared__ float sW[EDGE_FEAT * EDGE_HID];   // 512 floats
    __shared__ float sb[EDGE_HID];
    int tid = threadIdx.x;
    for (int i = tid; i < EDGE_FEAT * EDGE_HID; i += 256) sW[i] = W_et[i];
    if (tid < EDGE_HID) sb[tid] = b_et[tid];
    __syncthreads();

    int e = blockIdx.x * 256 + tid;
    if (e >= n_edges) return;

    int d = dst[e];
    float dn = denom[d];
    float alpha = exbuf[e] / (dn == 0.0f ? 1.0f : dn);

    float f[EDGE_FEAT];
    const float4* fp = (const float4*)(edge_feats + (size_t)e * EDGE_FEAT);
#pragma unroll
    for (int q = 0; q < 4; ++q) {
        float4 v = fp[q];
        f[q * 4 + 0] = v.x; f[q * 4 + 1] = v.y; f[q * 4 + 2] = v.z; f[q * 4 + 3] = v.w;
    }

    float o[EDGE_HID];
#pragma unroll
    for (int j = 0; j < EDGE_HID; ++j) o[j] = sb[j];
#pragma unroll
    for (int k = 0; k < EDGE_FEAT; ++k) {
        float fk = f[k];
#pragma unroll
        for (int j = 0; j < EDGE_HID; ++j) o[j] = fmaf(fk, sW[k * EDGE_HID + j], o[j]);
    }

    float* crow = c + (size_t)d * EDGE_HID;
#pragma unroll
    for (int j = 0; j < EDGE_HID; ++j) atomicAdd(&crow[j], alpha * o[j]);
}

// ---------------------------------------------------------------
// Kernel 5: node MLP with V_WMMA_F32_16X16X4_F32
// block = 32 nodes (2 M-tiles), 8 waves; wave w owns columns [16w,16w+16).
// Each B fragment is loaded once and reused for 2 WMMAs (M-tiles 0 and 1).
// GEMM1: (32x160)@(160x128)+b1, relu -> LDS
// GEMM2: (32x128)@(128x128)+b2, relu -> out
// ---------------------------------------------------------------
#define S1 164   // padded stride for h_in  (164 mod 64 = 36, gcd=4 -> conflict-free)
#define S2 132   // padded stride for h_mid (132 mod 64 = 4)

__global__ __launch_bounds__(256) void k_node_mlp(const float* __restrict__ c,
                                                  const float* __restrict__ node_feats,
                                                  const float* __restrict__ W1,
                                                  const float* __restrict__ b1,
                                                  const float* __restrict__ W2,
                                                  const float* __restrict__ b2,
                                                  float* __restrict__ out) {
    __shared__ float h_in[M_TILE * S1];    // 32*164*4 = 20.5 KB
    __shared__ float h_mid[M_TILE * S2];   // 32*132*4 = 16.5 KB

    const int node0 = blockIdx.x * M_TILE;
    const int tid = threadIdx.x;

    // stage h = [elu(c) , node_feats] for 32 nodes
    for (int idx = tid; idx < M_TILE * H_IN_DIM; idx += 256) {
        int r = idx / H_IN_DIM, col = idx % H_IN_DIM;
        float v;
        if (col < EDGE_HID) {
            float x = c[(size_t)(node0 + r) * EDGE_HID + col];
            v = (x > 0.0f) ? x : (expf(x) - 1.0f);            // elu
        } else {
            v = node_feats[(size_t)(node0 + r) * NODE_FEAT + (col - EDGE_HID)];
        }
        h_in[r * S1 + col] = v;
    }
    __syncthreads();

    const int wave = tid >> 5;          // 0..7 -> N tile
    const int lane = tid & 31;
    const int half = lane >> 4;         // 0/1
    const int li   = lane & 15;         // row / col index within tile
    const int n0   = wave * 16;

    // ---- GEMM1: K = 160 ----
    v8f acc0 = {}, acc1 = {};
    for (int k = 0; k < H_IN_DIM; k += 4) {
        v2f a0, a1, b;
        a0.x = h_in[li * S1 + k + 2 * half];
        a0.y = h_in[li * S1 + k + 2 * half + 1];
        a1.x = h_in[(li + 16) * S1 + k + 2 * half];
        a1.y = h_in[(li + 16) * S1 + k + 2 * half + 1];
        b.x  = W1[(size_t)(k + 2 * half) * NODE_FEAT + n0 + li];
        b.y  = W1[(size_t)(k + 2 * half + 1) * NODE_FEAT + n0 + li];
        acc0 = __builtin_amdgcn_wmma_f32_16x16x4_f32(false, a0, false, b,
                                                     (short)0, acc0, false, false);
        acc1 = __builtin_amdgcn_wmma_f32_16x16x4_f32(false, a1, false, b,
                                                     (short)0, acc1, false, false);
    }
    {
        float bias = b1[n0 + li];
#pragma unroll
        for (int r = 0; r < 8; ++r) {
            float v0 = acc0[r] + bias;
            float v1 = acc1[r] + bias;
            h_mid[(r + 8 * half) * S2 + n0 + li]        = (v0 > 0.0f) ? v0 : 0.0f;
            h_mid[(16 + r + 8 * half) * S2 + n0 + li]   = (v1 > 0.0f) ? v1 : 0.0f;
        }
    }
    __syncthreads();

    // ---- GEMM2: K = 128 ----
    v8f acc2a = {}, acc2b = {};
    for (int k = 0; k < NODE_FEAT; k += 4) {
        v2f a0, a1, b;
        a0.x = h_mid[li * S2 + k + 2 * half];
        a0.y = h_mid[li * S2 + k + 2 * half + 1];
        a1.x = h_mid[(li + 16) * S2 + k + 2 * half];
        a1.y = h_mid[(li + 16) * S2 + k + 2 * half + 1];
        b.x  = W2[(size_t)(k + 2 * half) * NODE_FEAT + n0 + li];
        b.y  = W2[(size_t)(k + 2 * half + 1) * NODE_FEAT + n0 + li];
        acc2a = __builtin_amdgcn_wmma_f32_16x16x4_f32(false, a0, false, b,
                                                      (short)0, acc2a, false, false);
        acc2b = __builtin_amdgcn_wmma_f32_16x16x4_f32(false, a1, false, b,
                                                      (short)0, acc2b, false, false);
    }
    {
        float bias = b2[n0 + li];
#pragma unroll
        for (int r = 0; r < 8; ++r) {
            float v0 = acc2a[r] + bias;
            float v1 = acc2b[r] + bias;
            out[(size_t)(node0 + r + 8 * half) * NODE_FEAT + n0 + li]      = (v0 > 0.0f) ? v0 : 0.0f;
            out[(size_t)(node0 + 16 + r + 8 * half) * NODE_FEAT + n0 + li] = (v1 > 0.0f) ? v1 : 0.0f;
        }
    }
}

// ---------------------------------------------------------------
static inline size_t align512(size_t x) { return (x + 511) & ~(size_t)511; }

extern "C" void kernel_launch(void* const* d_in, const int* in_sizes, int n_in,
                              void* d_out, int out_size, void* d_ws, size_t ws_size,
                              hipStream_t stream) {
    const float* edge_logits = (const float*)d_in[0];
    const float* edge_feats  = (const float*)d_in[1];
    const float* node_feats  = (const float*)d_in[2];
    const int*   dst         = (const int*)  d_in[3];
    const float* W_et        = (const float*)d_in[4];
    const float* b_et        = (const float*)d_in[5];
    const float* W1          = (const float*)d_in[6];
    const float* b1          = (const float*)d_in[7];
    const float* W2          = (const float*)d_in[8];
    const float* b2          = (const float*)d_in[9];
    float* out = (float*)d_out;

    const int n_edges = in_sizes[3];             // == N_EDGES
    (void)ws_size; (void)n_in; (void)out_size;

    // workspace layout
    char* ws = (char*)d_ws;
    size_t off = 0;
    unsigned* seg_max = (unsigned*)(ws + off); off = align512(off + (size_t)N_NODES * 4);
    float*    denom   = (float*)   (ws + off); off = align512(off + (size_t)N_NODES * 4);
    float*    exbuf   = (float*)   (ws + off); off = align512(off + (size_t)N_EDGES * 4);
    float*    cbuf    = (float*)   (ws + off); off = align512(off + (size_t)N_NODES * EDGE_HID * 4);

    const int edge_blocks = (n_edges + 255) / 256;
    const int init_blocks = (N_NODES * EDGE_HID + 255) / 256;
    const int node_blocks = N_NODES / M_TILE;    // 3125, exact

    k_init<<<init_blocks, 256, 0, stream>>>(seg_max, denom, cbuf);
    k_segmax<<<edge_blocks, 256, 0, stream>>>(edge_logits, dst, seg_max, n_edges);
    k_expsum<<<edge_blocks, 256, 0, stream>>>(edge_logits, dst, seg_max, exbuf, denom, n_edges);
    k_edge_xform<<<edge_blocks, 256, 0, stream>>>(edge_feats, exbuf, denom, dst,
                                                  W_et, b_et, cbuf, n_edges);
    k_node_mlp<<<node_blocks, 256, 0, stream>>>(cbuf, node_feats, W1, b1, W2, b2, out);
}